// BPNN_41051297415626
// MI455X (gfx1250) — compile-verified
//
#include <hip/hip_runtime.h>

typedef __attribute__((ext_vector_type(2))) float v2f;
typedef __attribute__((ext_vector_type(4))) float v4f;
typedef __attribute__((ext_vector_type(8))) float v8f;

#define NS   28
#define FEA  2048
#define HID  1024

// ---------------- init: zero de_dD accumulator + outputs ----------------
__global__ __launch_bounds__(256) void bpnn_init(float* __restrict__ de_dD,
                                                 float* __restrict__ out) {
  int i = blockIdx.x * 256 + threadIdx.x;
  if (i < NS * FEA) de_dD[i] = 0.f;
  if (i < 1 + NS * 3) out[i] = 0.f;
}

// ---------------- fused forward + dE/dDes, one streaming pass over W1 ----
// grid: (16 hblocks, 28 species), block: 128 threads = 4 waves.
// Each wave owns 16 hidden rows; WMMA f32 16x16x4 computes 16 row-dots.
__global__ __launch_bounds__(128) void bpnn_fused(
    const float* __restrict__ des, const float* __restrict__ W1,
    const float* __restrict__ b1,  const float* __restrict__ W2,
    float* __restrict__ de_dD, float* __restrict__ out) {
  __shared__ float desL[FEA];          // 8 KB   : descriptor for this species
  __shared__ float partial[4][FEA];    // 32 KB  : per-wave de_dD partials
  __shared__ float gbuf[4][16];        // per-wave relu'(z)*W2 for 16 rows
  __shared__ float eaccL;              // block energy accumulator

  const int s    = blockIdx.y;
  const int tid  = threadIdx.x;
  const int wid  = tid >> 5;
  const int lane = tid & 31;
  const int h0   = blockIdx.x * 64 + wid * 16;   // this wave's 16 rows

  if (tid == 0) eaccL = 0.f;
  const v4f* des4 = (const v4f*)(des + (size_t)s * FEA);
  for (int i = tid; i < FEA / 4; i += 128) ((v4f*)desL)[i] = des4[i];
  __syncthreads();

  // ---- Pass 1: z[h0..h0+15] via V_WMMA_F32_16X16X4_F32 ----
  // A 16x4 f32 layout: lanes 0-15 -> M=lane, VGPR0=K0,VGPR1=K1;
  //                    lanes 16-31 -> M=lane-16, VGPR0=K2,VGPR1=K3.
  const int hi = lane >> 4;            // 0: K={0,1}, 1: K={2,3}
  const int ml = lane & 15;            // M (row within 16-row block)
  const float* rowp = W1 + ((size_t)(s * HID + h0 + ml)) * FEA + 2 * hi;
  const float* desp = desL + 2 * hi;   // B: same des chunk in every column

  v8f c0 = {}, c1 = {}, c2 = {}, c3 = {};
  for (int f0 = 0; f0 < FEA; f0 += 16) {
    v2f a0 = *(const v2f*)(rowp + f0);
    v2f bb0 = *(const v2f*)(desp + f0);
    v2f a1 = *(const v2f*)(rowp + f0 + 4);
    v2f bb1 = *(const v2f*)(desp + f0 + 4);
    v2f a2 = *(const v2f*)(rowp + f0 + 8);
    v2f bb2 = *(const v2f*)(desp + f0 + 8);
    v2f a3 = *(const v2f*)(rowp + f0 + 12);
    v2f bb3 = *(const v2f*)(desp + f0 + 12);
    c0 = __builtin_amdgcn_wmma_f32_16x16x4_f32(false, a0, false, bb0, (short)0, c0, false, false);
    c1 = __builtin_amdgcn_wmma_f32_16x16x4_f32(false, a1, false, bb1, (short)0, c1, false, false);
    c2 = __builtin_amdgcn_wmma_f32_16x16x4_f32(false, a2, false, bb2, (short)0, c2, false, false);
    c3 = __builtin_amdgcn_wmma_f32_16x16x4_f32(false, a3, false, bb3, (short)0, c3, false, false);
  }
  c0 += c1; c2 += c3; c0 += c2;

  // C layout: VGPR r = row r (lanes 0-15) / row r+8 (lanes 16-31); all
  // columns identical -> lane 0 extracts rows 0-7, lane 16 rows 8-15.
  if (ml == 0) {
    const int mb = hi * 8;
    float e = 0.f;
    #pragma unroll
    for (int r = 0; r < 8; ++r) {
      const int h  = h0 + mb + r;
      const float z  = c0[r] + b1[s * HID + h];
      const float w2 = W2[s * HID + h];
      const float g  = (z > 0.f) ? w2 : 0.f;     // relu'(z) * W2
      gbuf[wid][mb + r] = g;
      e += (z > 0.f) ? z * w2 : 0.f;             // relu(z) * W2
    }
    atomicAdd(&eaccL, e);                        // ds_add_f32
  }
  __syncthreads();

  // ---- Pass 2: de_dD += sum_h g[h] * W1[h,:]  (rows still hot in L1/L2) --
  float g[16];
  #pragma unroll
  for (int m = 0; m < 16; ++m) g[m] = gbuf[wid][m];
  const float* bb = W1 + ((size_t)(s * HID + h0)) * FEA;
  for (int f0 = 0; f0 < FEA; f0 += 128) {
    const int fl = f0 + lane * 4;
    v4f acc = {};
    #pragma unroll
    for (int m = 0; m < 16; ++m) {
      v4f w = __builtin_nontemporal_load((const v4f*)(bb + (size_t)m * FEA + fl));
      acc += g[m] * w;
    }
    *(v4f*)&partial[wid][fl] = acc;
  }
  __syncthreads();

  // cross-wave reduce then one atomic pass per workgroup
  float* dd = de_dD + (size_t)s * FEA;
  for (int i = tid; i < FEA / 4; i += 128) {
    v4f sum = ((v4f*)partial[0])[i] + ((v4f*)partial[1])[i] +
              ((v4f*)partial[2])[i] + ((v4f*)partial[3])[i];
    atomicAdd(&dd[4 * i + 0], sum.x);
    atomicAdd(&dd[4 * i + 1], sum.y);
    atomicAdd(&dd[4 * i + 2], sum.z);
    atomicAdd(&dd[4 * i + 3], sum.w);
  }
  if (tid == 0) atomicAdd(out, eaccL);
}

// ---------------- force: F[jk] = -sum_s der[s,jk,:] . de_dD[s,:] ---------
__global__ __launch_bounds__(256) void bpnn_force(
    const float* __restrict__ der, const float* __restrict__ de_dD,
    const float* __restrict__ b2, float* __restrict__ out) {
  __shared__ float red[256];
  const int jk  = blockIdx.x;        // j*3+k, 0..83
  const int tid = threadIdx.x;
  float acc = 0.f;
  for (int s = 0; s < NS; ++s) {
    const v4f* dr = (const v4f*)(der + ((size_t)s * 84 + jk) * FEA);
    const v4f* dd = (const v4f*)(de_dD + (size_t)s * FEA);
    for (int i = tid; i < FEA / 4; i += 256) {
      v4f a = __builtin_nontemporal_load(dr + i);
      v4f b = dd[i];
      acc += a.x * b.x + a.y * b.y + a.z * b.z + a.w * b.w;
    }
  }
  red[tid] = acc;
  __syncthreads();
  for (int off = 128; off > 0; off >>= 1) {
    if (tid < off) red[tid] += red[tid + off];
    __syncthreads();
  }
  if (tid == 0) {
    out[1 + jk] = -red[0];
    if (jk == 0) {                    // fold in sum_s b2[s] (runs after fused)
      float bs = 0.f;
      for (int s = 0; s < NS; ++s) bs += b2[s];
      atomicAdd(out, bs);
    }
  }
}

extern "C" void kernel_launch(void* const* d_in, const int* in_sizes, int n_in,
                              void* d_out, int out_size, void* d_ws, size_t ws_size,
                              hipStream_t stream) {
  const float* des = (const float*)d_in[0];   // [28, 2048]
  const float* der = (const float*)d_in[1];   // [28, 28, 3, 2048]
  const float* W1  = (const float*)d_in[2];   // [28, 1024, 2048]
  const float* b1  = (const float*)d_in[3];   // [28, 1024]
  const float* W2  = (const float*)d_in[4];   // [28, 1, 1024]
  const float* b2  = (const float*)d_in[5];   // [28, 1]
  float* out   = (float*)d_out;               // [1 energy] + [28*3 force]
  float* de_dD = (float*)d_ws;                // 28*2048 floats = 229 KB

  bpnn_init<<<(NS * FEA + 255) / 256, 256, 0, stream>>>(de_dD, out);
  bpnn_fused<<<dim3(16, NS), 128, 0, stream>>>(des, W1, b1, W2, de_dD, out);
  bpnn_force<<<84, 256, 0, stream>>>(der, de_dD, b2, out);
}